// RF_Attn_17334488006716
// MI455X (gfx1250) — compile-verified
//
#include <hip/hip_runtime.h>

typedef float v2f __attribute__((ext_vector_type(2)));
typedef float v8f __attribute__((ext_vector_type(8)));

#define B_    16
#define CIN   256
#define C3    768
#define CCAT  512
#define COUT  256
#define HW    4096
#define WID   64

// ---------------------------------------------------------------------------
// GEMM1: qkv[b][o][hw] = sum_c w_qkv[o][c] * x[b][c][hw]
// M=768, K=256, N=4096 per batch.  V_WMMA_F32_16X16X4_F32.
// Block = 256 threads = 8 waves (4 M x 2 N); wave tile = 32x32 (2x2 WMMA tiles).
// Block tile = 128(M) x 64(N).  Grid: (768/128, 4096/64, 16).
// ---------------------------------------------------------------------------
__global__ __launch_bounds__(256) void k_gemm_qkv(
    const float* __restrict__ X,   // (B, CIN, HW)
    const float* __restrict__ Wq,  // (C3, CIN)
    float* __restrict__ Q)         // (B, C3, HW)
{
  const int b    = blockIdx.z;
  const int mBlk = blockIdx.x * 128;
  const int nBlk = blockIdx.y * 64;
  const int wave = threadIdx.x >> 5;
  const int lane = threadIdx.x & 31;
  const int half = lane >> 4;    // 0 or 1 (upper lanes hold K+2,K+3)
  const int idx  = lane & 15;    // row (A) / col (B)
  const int m0   = mBlk + (wave & 3) * 32;
  const int n0   = nBlk + (wave >> 2) * 32;

  const float* Xb = X + (size_t)b * CIN * HW;
  v8f acc[2][2] = {};

  for (int k0 = 0; k0 < CIN; k0 += 4) {
    v2f a[2], bf[2];
#pragma unroll
    for (int i = 0; i < 2; ++i) {
      // A 16x4 layout: lane m=idx, VGPR0=K(2*half), VGPR1=K(2*half+1) -> contiguous pair
      const float* p = Wq + (size_t)(m0 + i * 16 + idx) * CIN + k0 + 2 * half;
      a[i] = *(const v2f*)p;
    }
#pragma unroll
    for (int j = 0; j < 2; ++j) {
      // B 4x16 layout: lane n=idx, VGPR0 = row k0+2*half, VGPR1 = row k0+2*half+1
      const float* p = Xb + (size_t)(k0 + 2 * half) * HW + (n0 + j * 16 + idx);
      v2f t = { p[0], p[HW] };
      bf[j] = t;
    }
#pragma unroll
    for (int i = 0; i < 2; ++i)
#pragma unroll
      for (int j = 0; j < 2; ++j)
        acc[i][j] = __builtin_amdgcn_wmma_f32_16x16x4_f32(
            false, a[i], false, bf[j], (short)0, acc[i][j], false, false);
  }

  float* Qb = Q + (size_t)b * C3 * HW;
#pragma unroll
  for (int i = 0; i < 2; ++i)
#pragma unroll
    for (int j = 0; j < 2; ++j)
#pragma unroll
      for (int r = 0; r < 8; ++r) {
        // C/D layout: VGPR r -> row m0 + half*8 + r, col idx
        int row = m0 + i * 16 + half * 8 + r;
        int col = n0 + j * 16 + idx;
        Qb[(size_t)row * HW + col] = acc[i][j][r];
      }
}

// ---------------------------------------------------------------------------
// Depthwise 5x5 SAME conv over 768 channels.  One thread per output pixel;
// channel is uniform per block so weights become scalar loads.
// ---------------------------------------------------------------------------
__global__ __launch_bounds__(256) void k_dwconv(
    const float* __restrict__ Q,   // (B, C3, HW)
    const float* __restrict__ Wd,  // (C3, 1, 5, 5)
    float* __restrict__ A)         // (B, C3, HW)
{
  const int b = blockIdx.z;
  const int c = blockIdx.y;
  const int pix = blockIdx.x * 256 + threadIdx.x;
  const int y = pix >> 6;
  const int x = pix & 63;
  const float* plane = Q + ((size_t)b * C3 + c) * HW;
  const float* w = Wd + (size_t)c * 25;
  float s = 0.f;
#pragma unroll
  for (int dy = -2; dy <= 2; ++dy) {
    int yy = y + dy;
    if (yy < 0 || yy > 63) continue;
#pragma unroll
    for (int dx = -2; dx <= 2; ++dx) {
      int xx = x + dx;
      if (xx < 0 || xx > 63) continue;
      s += plane[yy * WID + xx] * w[(dy + 2) * 5 + (dx + 2)];
    }
  }
  A[((size_t)b * C3 + c) * HW + pix] = s;
}

// ---------------------------------------------------------------------------
// Per-(b,c) reductions over the 4096 pixels:
//   sum_k[b,c]  = sum relu(k);   sum_kv[b,c] = sum_k^2 * sum(v)
// k channel c: c<256 -> qkv[256+c], else agg[c]
// v channel c: c<256 -> qkv[512+c], else agg[256+c]
// ---------------------------------------------------------------------------
__global__ __launch_bounds__(256) void k_reduce_kv(
    const float* __restrict__ Q, const float* __restrict__ A,
    float* __restrict__ sum_k, float* __restrict__ sum_kv)
{
  const int bc = blockIdx.x;        // b*512 + c
  const int b = bc >> 9;
  const int c = bc & 511;
  const float* kp = (c < 256) ? Q + ((size_t)b * C3 + 256 + c) * HW
                              : A + ((size_t)b * C3 + c) * HW;
  const float* vp = (c < 256) ? Q + ((size_t)b * C3 + 512 + c) * HW
                              : A + ((size_t)b * C3 + 256 + c) * HW;
  float sk = 0.f, sv = 0.f;
  for (int i = threadIdx.x; i < HW; i += 256) {
    sk += fmaxf(kp[i], 0.f);
    sv += vp[i];
  }
  __shared__ float shk[256], shv[256];
  shk[threadIdx.x] = sk;
  shv[threadIdx.x] = sv;
  __syncthreads();
  for (int off = 128; off > 0; off >>= 1) {
    if (threadIdx.x < off) {
      shk[threadIdx.x] += shk[threadIdx.x + off];
      shv[threadIdx.x] += shv[threadIdx.x + off];
    }
    __syncthreads();
  }
  if (threadIdx.x == 0) {
    float K = shk[0], V = shv[0];
    sum_k[bc]  = K;
    sum_kv[bc] = K * K * V;
  }
}

// ---------------------------------------------------------------------------
// sum_q[b,hw] = sum over 512 q-channels of relu(q).  One thread per pixel.
// q channel c: c<256 -> qkv[c], else agg[c-256]
// ---------------------------------------------------------------------------
__global__ __launch_bounds__(256) void k_reduce_q(
    const float* __restrict__ Q, const float* __restrict__ A,
    float* __restrict__ sum_q)
{
  const int t  = blockIdx.x * 256 + threadIdx.x;  // b*HW + hw
  const int b  = t >> 12;
  const int hw = t & 4095;
  const float* q1 = Q + (size_t)b * C3 * HW + hw;
  const float* q2 = A + (size_t)b * C3 * HW + hw;
  float s = 0.f;
  for (int c = 0; c < 256; ++c) {
    s += fmaxf(q1[(size_t)c * HW], 0.f);
    s += fmaxf(q2[(size_t)c * HW], 0.f);
  }
  sum_q[t] = s;
}

// ---------------------------------------------------------------------------
// att[b,c,hw] = relu(q) * sum_kv[b,c] / (sum_q[b,hw] * sum_k[b,c] + 1e-15)
// ---------------------------------------------------------------------------
__global__ __launch_bounds__(256) void k_att(
    const float* __restrict__ Q, const float* __restrict__ A,
    const float* __restrict__ sum_k, const float* __restrict__ sum_kv,
    const float* __restrict__ sum_q, float* __restrict__ ATT)
{
  const size_t t = (size_t)blockIdx.x * 256 + threadIdx.x;  // B*512*HW
  const int b  = (int)(t >> 21);
  const int c  = (int)((t >> 12) & 511);
  const int hw = (int)(t & 4095);
  float qv = (c < 256) ? Q[((size_t)b * C3 + c) * HW + hw]
                       : A[((size_t)b * C3 + (c - 256)) * HW + hw];
  qv = fmaxf(qv, 0.f);
  const int bc = b * 512 + c;
  ATT[t] = qv * sum_kv[bc] / (sum_q[b * HW + hw] * sum_k[bc] + 1e-15f);
}

// ---------------------------------------------------------------------------
// GEMM2 + BN: out[b][o][hw] = BN( sum_c w_proj[o][c] * att[b][c][hw] )
// M=256, K=512, N=4096 per batch.  Same WMMA structure as GEMM1.
// ---------------------------------------------------------------------------
__global__ __launch_bounds__(256) void k_gemm_proj_bn(
    const float* __restrict__ ATT,  // (B, CCAT, HW)
    const float* __restrict__ Wp,   // (COUT, CCAT)
    const float* __restrict__ g, const float* __restrict__ be,
    const float* __restrict__ mu, const float* __restrict__ va,
    float* __restrict__ OUT)        // (B, COUT, HW)
{
  const int b    = blockIdx.z;
  const int mBlk = blockIdx.x * 128;
  const int nBlk = blockIdx.y * 64;
  const int wave = threadIdx.x >> 5;
  const int lane = threadIdx.x & 31;
  const int half = lane >> 4;
  const int idx  = lane & 15;
  const int m0   = mBlk + (wave & 3) * 32;
  const int n0   = nBlk + (wave >> 2) * 32;

  const float* Ab = ATT + (size_t)b * CCAT * HW;
  v8f acc[2][2] = {};

  for (int k0 = 0; k0 < CCAT; k0 += 4) {
    v2f a[2], bf[2];
#pragma unroll
    for (int i = 0; i < 2; ++i) {
      const float* p = Wp + (size_t)(m0 + i * 16 + idx) * CCAT + k0 + 2 * half;
      a[i] = *(const v2f*)p;
    }
#pragma unroll
    for (int j = 0; j < 2; ++j) {
      const float* p = Ab + (size_t)(k0 + 2 * half) * HW + (n0 + j * 16 + idx);
      v2f t = { p[0], p[HW] };
      bf[j] = t;
    }
#pragma unroll
    for (int i = 0; i < 2; ++i)
#pragma unroll
      for (int j = 0; j < 2; ++j)
        acc[i][j] = __builtin_amdgcn_wmma_f32_16x16x4_f32(
            false, a[i], false, bf[j], (short)0, acc[i][j], false, false);
  }

  float* Ob = OUT + (size_t)b * COUT * HW;
#pragma unroll
  for (int i = 0; i < 2; ++i)
#pragma unroll
    for (int r = 0; r < 8; ++r) {
      const int row = m0 + i * 16 + half * 8 + r;
      const float sc = g[row] * rsqrtf(va[row] + 1e-5f);
      const float mn = mu[row];
      const float bt = be[row];
#pragma unroll
      for (int j = 0; j < 2; ++j) {
        const int col = n0 + j * 16 + idx;
        Ob[(size_t)row * HW + col] = (acc[i][j][r] - mn) * sc + bt;
      }
    }
}

// ---------------------------------------------------------------------------
extern "C" void kernel_launch(void* const* d_in, const int* in_sizes, int n_in,
                              void* d_out, int out_size, void* d_ws, size_t ws_size,
                              hipStream_t stream) {
  (void)in_sizes; (void)n_in; (void)out_size; (void)ws_size;
  const float* x      = (const float*)d_in[0];
  const float* w_qkv  = (const float*)d_in[1];
  const float* w_dw   = (const float*)d_in[2];
  const float* w_proj = (const float*)d_in[3];
  const float* gamma  = (const float*)d_in[4];
  const float* beta   = (const float*)d_in[5];
  const float* mean   = (const float*)d_in[6];
  const float* var    = (const float*)d_in[7];
  float* out = (float*)d_out;

  // Workspace layout (floats): qkv | agg | att | sum_k | sum_kv | sum_q
  float* ws   = (float*)d_ws;
  float* qkv  = ws;                                   // 16*768*4096
  float* agg  = qkv + (size_t)B_ * C3 * HW;           // 16*768*4096
  float* att  = agg + (size_t)B_ * C3 * HW;           // 16*512*4096
  float* s_k  = att + (size_t)B_ * CCAT * HW;         // 16*512
  float* s_kv = s_k + (size_t)B_ * CCAT;              // 16*512
  float* s_q  = s_kv + (size_t)B_ * CCAT;             // 16*4096

  k_gemm_qkv<<<dim3(C3 / 128, HW / 64, B_), 256, 0, stream>>>(x, w_qkv, qkv);
  k_dwconv<<<dim3(HW / 256, C3, B_), 256, 0, stream>>>(qkv, w_dw, agg);
  k_reduce_kv<<<B_ * CCAT, 256, 0, stream>>>(qkv, agg, s_k, s_kv);
  k_reduce_q<<<(B_ * HW) / 256, 256, 0, stream>>>(qkv, agg, s_q);
  k_att<<<(B_ * CCAT * HW) / 256, 256, 0, stream>>>(qkv, agg, s_k, s_kv, s_q, att);
  k_gemm_proj_bn<<<dim3(COUT / 128, HW / 64, B_), 256, 0, stream>>>(
      att, w_proj, gamma, beta, mean, var, out);
}